// AdaptiveComputationTime_7645041787087
// MI455X (gfx1250) — compile-verified
//
#include <hip/hip_runtime.h>
#include <hip/hip_bf16.h>
#include <stdint.h>

#define B_ 32
#define M_ 2048
#define H_ 1024
#define THRESH 0.99f

typedef float v2f __attribute__((ext_vector_type(2)));
typedef float v8f __attribute__((ext_vector_type(8)));

// ---------------- Kernel 1: rank = clip(cumsum(run)-1, 0) per batch ----------------
__global__ void act_rank_kernel(const unsigned char* __restrict__ run,
                                int* __restrict__ idx) {
  int b = blockIdx.x;
  int lane = threadIdx.x;               // wave32
  unsigned ltmask = (1u << lane) - 1u;
  const unsigned char* rb = run + (size_t)b * M_;
  int* ib = idx + (size_t)b * M_;
  int base = 0;
  for (int i = 0; i < M_; i += 32) {
    int m = i + lane;
    int r = rb[m] ? 1 : 0;
    unsigned ball = (unsigned)__ballot(r);
    int pre = __popc(ball & ltmask);
    int rank = base + pre + r - 1;      // inclusive cumsum - 1
    ib[m] = rank < 0 ? 0 : rank;        // clip low (high clip impossible)
    base += __popc(ball);
  }
}

// ---------------- Kernel 2: WMMA GEMV + ACT update + weighted blend ----------------
// One wave per 16 tokens. D = A(16x4 f32) x B(4x16 f32) + C accumulated over H/4 steps.
// B-fragment replicates the W chunk into all 16 columns -> every D column = per-token dot.
__global__ void act_main_kernel(const float* __restrict__ h,
                                const float* __restrict__ W,
                                const float* __restrict__ bias,
                                const float* __restrict__ weighted_h,
                                const float* __restrict__ acc_p,
                                const unsigned char* __restrict__ run,
                                const int* __restrict__ idx,
                                int* __restrict__ run_new,
                                float* __restrict__ out_weighted,
                                float* __restrict__ out_acc) {
  __shared__ float dots[16];
  __shared__ float padj_s[16];
  __shared__ int   srcs[16];

  int tile = blockIdx.x;
  int b  = tile >> 7;              // / (M/16)
  int m0 = (tile & 127) << 4;      // * 16
  int lane = threadIdx.x;
  int tok  = lane & 15;            // A: lanes 0-15 and 16-31 both map M=0..15
  int hi   = lane >> 4;            // 0 -> K={0,1}, 1 -> K={2,3} (A and B fragments alike)

  size_t bm0 = (size_t)b * M_ + m0;
  int src = idx[bm0 + tok];
  const float* hrow = h + ((size_t)b * M_ + src) * (size_t)H_;

  v8f c0 = {0.f,0.f,0.f,0.f,0.f,0.f,0.f,0.f};
  v8f c1 = {0.f,0.f,0.f,0.f,0.f,0.f,0.f,0.f};
  const float* hk = hrow + 2 * hi;
  const float* wk = W    + 2 * hi;
#pragma unroll 4
  for (int k = 0; k < H_; k += 8) {
    v2f a0 = *(const v2f*)(hk + k);
    v2f w0 = *(const v2f*)(wk + k);
    v2f a1 = *(const v2f*)(hk + k + 4);
    v2f w1 = *(const v2f*)(wk + k + 4);
    c0 = __builtin_amdgcn_wmma_f32_16x16x4_f32(false, a0, false, w0, (short)0, c0, false, false);
    c1 = __builtin_amdgcn_wmma_f32_16x16x4_f32(false, a1, false, w1, (short)0, c1, false, false);
  }

  // D layout: c[i] on lane 0 = dot(token i), on lane 16 = dot(token 8+i)
  if (lane == 0) {
#pragma unroll
    for (int i = 0; i < 8; ++i) dots[i] = c0[i] + c1[i];
  } else if (lane == 16) {
#pragma unroll
    for (int i = 0; i < 8; ++i) dots[8 + i] = c0[i] + c1[i];
  }
  if (hi == 0) srcs[tok] = src;
  __syncthreads();

  if (lane < 16) {
    size_t gm = bm0 + lane;
    int   runm = run[gm] ? 1 : 0;
    float x    = dots[lane] + bias[0];
    float p    = runm ? (1.0f / (1.0f + __expf(-x))) : 0.0f;
    float accp = acc_p[gm];
    float accn = accp + p;
    int cont = (accn < THRESH) && runm;
    int ext  = runm && !cont;
    float padj = cont ? p : (ext ? (1.0f - accp) : 0.0f);
    out_acc[gm]  = accn;
    run_new[gm]  = cont;
    padj_s[lane] = padj;
  }
  __syncthreads();

  // weighted_new = h_un * p_adj + weighted_h * (1 - p_adj), float4-coalesced
  for (int t = 0; t < 16; ++t) {
    float padj = padj_s[t];
    size_t gm = bm0 + t;
    const float4* wr = (const float4*)(weighted_h + gm * (size_t)H_);
    float4*       wo = (float4*)(out_weighted + gm * (size_t)H_);
    if (padj != 0.0f) {
      const float4* hr = (const float4*)(h + ((size_t)b * M_ + srcs[t]) * (size_t)H_);
      float q = 1.0f - padj;
#pragma unroll
      for (int i = 0; i < 8; ++i) {
        int col = lane + 32 * i;
        float4 hv = hr[col], wv = wr[col], o;
        o.x = hv.x * padj + wv.x * q;
        o.y = hv.y * padj + wv.y * q;
        o.z = hv.z * padj + wv.z * q;
        o.w = hv.w * padj + wv.w * q;
        wo[col] = o;
      }
    } else {
#pragma unroll
      for (int i = 0; i < 8; ++i) { int col = lane + 32 * i; wo[col] = wr[col]; }
    }
  }
}

// ---------------- Kernel 3: compaction scan; compose pack gather with unpack gather ----
__global__ void act_newpos_kernel(const int* __restrict__ run_new,
                                  const int* __restrict__ idx,
                                  int* __restrict__ srcof,
                                  int* __restrict__ count) {
  int b = blockIdx.x;
  int lane = threadIdx.x;
  unsigned ltmask = (1u << lane) - 1u;
  const int* rb = run_new + (size_t)b * M_;
  const int* ib = idx + (size_t)b * M_;
  int* sb = srcof + (size_t)b * M_;
  int base = 0;
  for (int i = 0; i < M_; i += 32) {
    int m = i + lane;
    int r = rb[m];
    unsigned ball = (unsigned)__ballot(r);
    int pre = __popc(ball & ltmask);
    if (r) sb[base + pre] = ib[m];     // source row in h for packed row (base+pre)
    base += __popc(ball);
  }
  if (lane == 0) count[b] = base;
}

// ---------------- Kernel 4: pack (gather rows of h, zero the tail) ----------------
__global__ void act_pack_kernel(const float* __restrict__ h,
                                const int* __restrict__ srcof,
                                const int* __restrict__ count,
                                float* __restrict__ out_packed) {
  int blk = blockIdx.x;
  int b = blk >> 11;                  // / M_
  int j = blk & (M_ - 1);
  int t = threadIdx.x;                // 256 threads * float4 = 1024 floats
  float4* dst = (float4*)(out_packed + ((size_t)b * M_ + j) * (size_t)H_);
  if (j < count[b]) {
    int src = srcof[(size_t)b * M_ + j];
    const float4* s = (const float4*)(h + ((size_t)b * M_ + src) * (size_t)H_);
    dst[t] = s[t];
  } else {
    float4 z = {0.f, 0.f, 0.f, 0.f};
    dst[t] = z;
  }
}

extern "C" void kernel_launch(void* const* d_in, const int* in_sizes, int n_in,
                              void* d_out, int out_size, void* d_ws, size_t ws_size,
                              hipStream_t stream) {
  const float*         h    = (const float*)d_in[0];
  const float*         W    = (const float*)d_in[1];
  const float*         bias = (const float*)d_in[2];
  const float*         wh   = (const float*)d_in[3];
  const float*         accp = (const float*)d_in[4];
  const unsigned char* run  = (const unsigned char*)d_in[5];

  float* out          = (float*)d_out;
  float* out_packed   = out;
  float* out_weighted = out + (size_t)B_ * M_ * H_;
  float* out_acc      = out + 2 * (size_t)B_ * M_ * H_;

  int* idx     = (int*)d_ws;
  int* run_new = idx + (size_t)B_ * M_;
  int* srcof   = run_new + (size_t)B_ * M_;
  int* count   = srcof + (size_t)B_ * M_;

  act_rank_kernel<<<B_, 32, 0, stream>>>(run, idx);
  act_main_kernel<<<B_ * M_ / 16, 32, 0, stream>>>(h, W, bias, wh, accp, run,
                                                   idx, run_new, out_weighted, out_acc);
  act_newpos_kernel<<<B_, 32, 0, stream>>>(run_new, idx, srcof, count);
  act_pack_kernel<<<B_ * M_, 256, 0, stream>>>(h, srcof, count, out_packed);
}